// VectorQuantizer_33165737459753
// MI455X (gfx1250) — compile-verified
//
#include <hip/hip_runtime.h>
#include <stdint.h>

typedef __attribute__((ext_vector_type(2))) float v2f;
typedef __attribute__((ext_vector_type(8))) float v8f;

#define NUM_EMB   512
#define DIM       64
#define N_ROWS    65536              // 64 * 32 * 32  (B*H*W)
#define Q_SIZE    (N_ROWS * DIM)     // 4,194,304
#define LOSS_OFF  Q_SIZE
#define IDX_OFF   (Q_SIZE + 1)
#define N_BLOCKS  (N_ROWS / 16)      // 4096

__device__ __forceinline__ unsigned long long packKey(float s, int col) {
    unsigned u = __float_as_uint(s);
    u = (u & 0x80000000u) ? ~u : (u | 0x80000000u);   // monotone order-preserving map
    return ((unsigned long long)u << 32) | (unsigned)col;
}

__device__ __forceinline__ unsigned long long umin64(unsigned long long a, unsigned long long b) {
    return a < b ? a : b;
}

__global__ __launch_bounds__(256) void vq_kernel(const float* __restrict__ x,
                                                 const float* __restrict__ emb,
                                                 float* __restrict__ out,
                                                 float* __restrict__ partial) {
    __shared__ float ldsX[16 * 65];              // padded: stride 65 -> conflict free
    __shared__ unsigned long long ldsMin[16];
    __shared__ int ldsIdx[16];
    __shared__ float ldsSum[256];

    const int wg = blockIdx.x;                   // 0..4095
    const int t  = threadIdx.x;                  // 0..255
    const int rowBase = wg * 16;
    const int bIdx = rowBase >> 10;              // batch index (all 16 rows share it)
    const int hwB  = rowBase & 1023;             // h*32+w base

    // ---- stage x tile [16 rows x 64 dims] NCHW -> LDS row-major ----
    {
        const int r = t & 15;
        const int dBase = t >> 4;
        const float* xb = x + (size_t)bIdx * 65536 + hwB + r;   // + d*1024 per dim
        #pragma unroll
        for (int p = 0; p < 4; ++p) {
            const int d = dBase + p * 16;
            ldsX[r * 65 + d] = xb[(size_t)d * 1024];
        }
    }
    if (t < 16) ldsMin[t] = ~0ull;
    __syncthreads();

    // ---- WMMA: dot(x_row, e_col) for 16 rows x 512 cols ----
    const int lane   = t & 31;
    const int wave   = t >> 5;                   // 0..7, each owns 64 codes
    const int laneHi = lane >> 4;                // half select
    const int laneLo = lane & 15;
    const int kofs   = laneHi * 2;               // A/B K sub-offset per half
    const int colB   = wave * 64;

    v8f acc0 = {}, acc1 = {}, acc2 = {}, acc3 = {};
    float en0 = 0.f, en1 = 0.f, en2 = 0.f, en3 = 0.f;

    const float* ax = &ldsX[laneLo * 65 + kofs];
    const float* e0 = emb + (size_t)(colB +  0 + laneLo) * DIM + kofs;
    const float* e1 = emb + (size_t)(colB + 16 + laneLo) * DIM + kofs;
    const float* e2 = emb + (size_t)(colB + 32 + laneLo) * DIM + kofs;
    const float* e3 = emb + (size_t)(colB + 48 + laneLo) * DIM + kofs;

    #pragma unroll
    for (int kk = 0; kk < DIM; kk += 4) {
        v2f a;  a.x  = ax[kk]; a.y  = ax[kk + 1];
        v2f b0; b0.x = e0[kk]; b0.y = e0[kk + 1];
        v2f b1; b1.x = e1[kk]; b1.y = e1[kk + 1];
        v2f b2; b2.x = e2[kk]; b2.y = e2[kk + 1];
        v2f b3; b3.x = e3[kk]; b3.y = e3[kk + 1];
        en0 += b0.x * b0.x + b0.y * b0.y;
        en1 += b1.x * b1.x + b1.y * b1.y;
        en2 += b2.x * b2.x + b2.y * b2.y;
        en3 += b3.x * b3.x + b3.y * b3.y;
        acc0 = __builtin_amdgcn_wmma_f32_16x16x4_f32(false, a, false, b0, (short)0, acc0, false, false);
        acc1 = __builtin_amdgcn_wmma_f32_16x16x4_f32(false, a, false, b1, (short)0, acc1, false, false);
        acc2 = __builtin_amdgcn_wmma_f32_16x16x4_f32(false, a, false, b2, (short)0, acc2, false, false);
        acc3 = __builtin_amdgcn_wmma_f32_16x16x4_f32(false, a, false, b3, (short)0, acc3, false, false);
    }

    // complete ||e_col||^2: partner lane (xor 16) holds complementary K subset, same col
    const float eN0 = en0 + __shfl_xor(en0, 16, 32);
    const float eN1 = en1 + __shfl_xor(en1, 16, 32);
    const float eN2 = en2 + __shfl_xor(en2, 16, 32);
    const float eN3 = en3 + __shfl_xor(en3, 16, 32);

    // ---- per-row argmin of score = ||e||^2 - 2*dot (||x||^2 constant per row) ----
    unsigned long long cand[8];
    #pragma unroll
    for (int i = 0; i < 8; ++i) {
        unsigned long long k0 = packKey(eN0 - 2.f * acc0[i], colB +  0 + laneLo);
        unsigned long long k1 = packKey(eN1 - 2.f * acc1[i], colB + 16 + laneLo);
        unsigned long long k2 = packKey(eN2 - 2.f * acc2[i], colB + 32 + laneLo);
        unsigned long long k3 = packKey(eN3 - 2.f * acc3[i], colB + 48 + laneLo);
        cand[i] = umin64(umin64(k0, k1), umin64(k2, k3));
    }
    #pragma unroll
    for (int m = 1; m < 16; m <<= 1) {       // reduce across 16 lanes of each half
        #pragma unroll
        for (int i = 0; i < 8; ++i) {
            unsigned long long o = __shfl_xor(cand[i], m, 32);
            cand[i] = umin64(cand[i], o);
        }
    }
    if (laneLo == 0) {                        // one lane per half: rows laneHi*8 + i
        #pragma unroll
        for (int i = 0; i < 8; ++i)
            atomicMin(&ldsMin[laneHi * 8 + i], cand[i]);
    }
    __syncthreads();

    if (t < 16) {
        const unsigned long long k = ldsMin[t];
        const int idx = (int)(unsigned)(k & 0xFFFFFFFFull);
        ldsIdx[t] = idx;
        out[IDX_OFF + rowBase + t] = (float)idx;
    }
    __syncthreads();

    // ---- gather quantized (NCHW) + loss partial ----
    float lsum = 0.f;
    {
        const int r = t & 15;
        const int dBase = t >> 4;
        const float* ev = emb + (size_t)ldsIdx[r] * DIM;
        float* oq = out + (size_t)bIdx * 65536 + hwB + r;
        #pragma unroll
        for (int p = 0; p < 4; ++p) {
            const int d = dBase + p * 16;
            const float q  = ev[d];
            const float xv = ldsX[r * 65 + d];
            const float df = q - xv;
            lsum += df * df;
            oq[(size_t)d * 1024] = q;
        }
    }
    ldsSum[t] = lsum;
    __syncthreads();
    #pragma unroll
    for (int n = 128; n > 0; n >>= 1) {
        if (t < n) ldsSum[t] += ldsSum[t + n];
        __syncthreads();
    }
    if (t == 0) partial[wg] = ldsSum[0];
}

__global__ __launch_bounds__(256) void vq_loss_reduce(const float* __restrict__ partial,
                                                      float* __restrict__ out) {
    __shared__ float s[256];
    float a = 0.f;
    for (int i = threadIdx.x; i < N_BLOCKS; i += 256) a += partial[i];
    s[threadIdx.x] = a;
    __syncthreads();
    #pragma unroll
    for (int n = 128; n > 0; n >>= 1) {
        if (threadIdx.x < n) s[threadIdx.x] += s[threadIdx.x + n];
        __syncthreads();
    }
    // loss = e_latent + 0.25*q_latent, forward-equal -> 1.25 * mean((q-x)^2)
    if (threadIdx.x == 0) out[LOSS_OFF] = s[0] * 1.25f / (float)Q_SIZE;
}

extern "C" void kernel_launch(void* const* d_in, const int* in_sizes, int n_in,
                              void* d_out, int out_size, void* d_ws, size_t ws_size,
                              hipStream_t stream) {
    const float* x   = (const float*)d_in[0];   // [64,64,32,32] f32 NCHW
    const float* emb = (const float*)d_in[1];   // [512,64] f32
    float* out = (float*)d_out;                 // [quantized | loss | idx] flat
    float* partial = (float*)d_ws;              // 4096 floats

    vq_kernel<<<N_BLOCKS, 256, 0, stream>>>(x, emb, out, partial);
    vq_loss_reduce<<<1, 256, 0, stream>>>(partial, out);
}